// KDAPolicyNetwork_77000173682738
// MI455X (gfx1250) — compile-verified
//
#include <hip/hip_runtime.h>

typedef float v2f __attribute__((ext_vector_type(2)));
typedef float v8f __attribute__((ext_vector_type(8)));

namespace {
constexpr int kT = 1024, kDKP = 32, kDV = 16, kN = 4;
constexpr float kEps = 1.1920929e-07f;  // torch float32 eps
}

// sum over the 16-lane half-groups (column dimension e); result uniform per group
__device__ __forceinline__ float red16(float x) {
  x += __shfl_xor(x, 1, 32);
  x += __shfl_xor(x, 2, 32);
  x += __shfl_xor(x, 4, 32);
  x += __shfl_xor(x, 8, 32);
  return x;
}

__device__ __forceinline__ float sigmoidf(float x) {
  return 1.0f / (1.0f + __expf(-x));
}

// One wave32 per batch. State S[4][16]: stream i, register r holds element
// (d = 4*(r>>1) + (r&1) + 2*(lane>=16), e = lane&15) of the 32x16 matrix —
// exactly the B-operand striping of V_WMMA_F32_16X16X4_F32 chunk r>>1.
// Row-striped q/k/alpha come straight from global memory as b64 pair loads
// (uniform-per-half addresses -> cache broadcast), software-pipelined one step
// ahead so the serial T-chain never waits on VMEM. Per-iteration DS traffic is
// two ds_bpermute ops total.
__global__ __launch_bounds__(32, 1)
void kda_mhc_wave_kernel(
    const float* __restrict__ qp, const float* __restrict__ kp,
    const float* __restrict__ vp, const float* __restrict__ ap,
    const float* __restrict__ bp, const float* __restrict__ S_init,
    const float* __restrict__ w_norm, const float* __restrict__ w_pre,
    const float* __restrict__ w_post, const float* __restrict__ w_res,
    const float* __restrict__ b_pre, const float* __restrict__ b_post,
    const float* __restrict__ b_res, const float* __restrict__ p_apre,
    const float* __restrict__ p_apost, const float* __restrict__ p_ares,
    float* __restrict__ out) {
  const int b = blockIdx.x;
  const int lane = threadIdx.x;
  const int l16 = lane & 15;
  const int hioff = (lane >> 4) << 1;  // 0 for lanes 0-15, 2 for lanes 16-31

  const float a_pre = p_apre[0], a_post = p_apost[0], a_res = p_ares[0];

  // ------------------- load initial state (registers) -------------------
  float S[4][16];
#pragma unroll
  for (int i = 0; i < 4; ++i)
#pragma unroll
    for (int r = 0; r < 16; ++r) {
      const int d = ((r >> 1) << 2) + (r & 1) + hioff;
      S[i][r] = S_init[((b * kN + i) * kDKP + d) * kDV + l16];
    }

  // ------------------- gates from S_init (once per wave) -------------------
  float mag[4];
#pragma unroll
  for (int i = 0; i < 4; ++i) {
    float p = 0.f;
#pragma unroll
    for (int r = 0; r < 16; ++r) p = fmaf(S[i][r], S[i][r], p);
    p += __shfl_xor(p, 16, 32);  // other wave-half holds the other 16 rows
    mag[i] = p * (1.0f / 16.0f);
  }
  float ss = 0.f;
#pragma unroll
  for (int i = 0; i < 4; ++i) ss = fmaf(mag[i], mag[i], ss);
  ss = red16(ss);
  const float rinv = rsqrtf(ss * (1.0f / 64.0f) + kEps);
  float xv[4];
#pragma unroll
  for (int i = 0; i < 4; ++i) xv[i] = mag[i] * rinv * w_norm[i * 16 + l16];

  float Hpre[4], Hpost[4], M[16];
#pragma unroll
  for (int o = 0; o < 4; ++o) {
    float p0 = 0.f, p1 = 0.f;
#pragma unroll
    for (int i = 0; i < 4; ++i) {
      p0 = fmaf(xv[i], w_pre[o * 64 + i * 16 + l16], p0);
      p1 = fmaf(xv[i], w_post[o * 64 + i * 16 + l16], p1);
    }
    Hpre[o] = sigmoidf(fmaf(a_pre, red16(p0), b_pre[o]));
    Hpost[o] = 2.0f * sigmoidf(fmaf(a_post, red16(p1), b_post[o]));
  }
#pragma unroll
  for (int o = 0; o < 16; ++o) {
    float p = 0.f;
#pragma unroll
    for (int i = 0; i < 4; ++i) p = fmaf(xv[i], w_res[o * 64 + i * 16 + l16], p);
    M[o] = __expf(fmaf(a_res, red16(p), b_res[o]));
  }
#pragma unroll
  for (int it = 0; it < 6; ++it) {
#pragma unroll
    for (int i = 0; i < 4; ++i) {
      const float rs = 1.0f / (M[i * 4] + M[i * 4 + 1] + M[i * 4 + 2] + M[i * 4 + 3]);
      M[i * 4] *= rs; M[i * 4 + 1] *= rs; M[i * 4 + 2] *= rs; M[i * 4 + 3] *= rs;
    }
#pragma unroll
    for (int j = 0; j < 4; ++j) {
      const float cs = 1.0f / (M[j] + M[4 + j] + M[8 + j] + M[12 + j]);
      M[j] *= cs; M[4 + j] *= cs; M[8 + j] *= cs; M[12 + j] *= cs;
    }
  }

  // ------------------- sequential recurrence over T -------------------
  const float* qbase = qp + (size_t)b * kT * kDKP;
  const float* kbase = kp + (size_t)b * kT * kDKP;
  const float* abase = ap + (size_t)b * kT * kDKP;
  const float* vbase = vp + (size_t)b * kT * kDV;
  const float* bbase = bp + (size_t)b * kT;
  float* orow = out + (size_t)b * kT * kDV + lane;

  // pipelined input registers: striped b64 pairs + per-lane/scalar values
  float2 kC[8], aC[8], qC[8];
  float vlC, btC;

  {
#pragma unroll
    for (int m = 0; m < 8; ++m) {
      kC[m] = *(const float2*)(kbase + 4 * m + hioff);
      aC[m] = *(const float2*)(abase + 4 * m + hioff);
      qC[m] = *(const float2*)(qbase + 4 * m + hioff);
    }
    vlC = vbase[l16];
    btC = bbase[0];
  }

#pragma unroll 1
  for (int t = 0; t < kT; ++t) {
    // ---- issue loads for t+1 first (hide VMEM latency behind this step) ----
    const int tn = (t + 1 < kT) ? (t + 1) : (kT - 1);
    float2 kNx[8], aNx[8], qNx[8];
    float vlN, btN;
    {
      const float* qr = qbase + tn * kDKP;
      const float* kr2 = kbase + tn * kDKP;
      const float* ar = abase + tn * kDKP;
#pragma unroll
      for (int m = 0; m < 8; ++m) {
        kNx[m] = *(const float2*)(kr2 + 4 * m + hioff);
        aNx[m] = *(const float2*)(ar + 4 * m + hioff);
        qNx[m] = *(const float2*)(qr + 4 * m + hioff);
      }
      vlN = vbase[tn * kDV + l16];
      btN = bbase[tn];
    }
    if (t + 16 < kT) {  // deep prefetch into L2/L0 (global_prefetch_b8)
      __builtin_prefetch(qbase + (t + 16) * kDKP + lane, 0, 1);
      __builtin_prefetch(kbase + (t + 16) * kDKP + lane, 0, 1);
      __builtin_prefetch(abase + (t + 16) * kDKP + lane, 0, 1);
    }

    // ---- q.k from the striped pairs (halves hold complementary d-sets) ----
    float qk = 0.f;
#pragma unroll
    for (int m = 0; m < 8; ++m) {
      qk = fmaf(qC[m].x, kC[m].x, qk);
      qk = fmaf(qC[m].y, kC[m].y, qk);
    }
    qk += __shfl_xor(qk, 16, 32);  // single cross-half combine

    // ---- pre-mix s = sum_i Hpre[i]*S[i] (decay folded in at WMMA build) ----
    float sr[16];
#pragma unroll
    for (int r = 0; r < 16; ++r) {
      float s = Hpre[0] * S[0][r];
      s = fmaf(Hpre[1], S[1][r], s);
      s = fmaf(Hpre[2], S[2][r], s);
      s = fmaf(Hpre[3], S[3][r], s);
      sr[r] = s;
    }

    // ---- four independent 2-deep WMMA chains over the 8 K-chunks ----
    // A row 0 = k_t, row 1 = q_t, rows 2..15 = 0; B = alpha(row) * s
    // => D row0 = k^T aS, D row1 = q^T aS
    v8f acc[4] = {{}, {}, {}, {}};
#pragma unroll
    for (int kc = 0; kc < 8; ++kc) {
      v2f A, Bv;
      A.x = (l16 == 0) ? kC[kc].x : ((l16 == 1) ? qC[kc].x : 0.0f);
      A.y = (l16 == 0) ? kC[kc].y : ((l16 == 1) ? qC[kc].y : 0.0f);
      Bv.x = aC[kc].x * sr[2 * kc];
      Bv.y = aC[kc].y * sr[2 * kc + 1];
      acc[kc & 3] = __builtin_amdgcn_wmma_f32_16x16x4_f32(
          false, A, false, Bv, (short)0, acc[kc & 3], false, false);
    }
    const float kt0 = (acc[0][0] + acc[1][0]) + (acc[2][0] + acc[3][0]);
    const float qt0 = (acc[0][1] + acc[1][1]) + (acc[2][1] + acc[3][1]);
    const float ktA = __shfl(kt0, l16, 32);  // (k^T aS)[e] broadcast to all lanes

    const float w_e = vlC - ktA;  // (v - k^T aS)[e]
    const float bw = btC * w_e;   // beta * w
    // out_t = q^T s_new = q^T aS + (q.k) * beta * w ; row1 of D lives on lanes 0-15
    if (lane < 16) orow[t * kDV] = fmaf(qk, bw, qt0);

    // ---- delta = (alpha-1)*s + beta*k (x) w ; residual-mix state update ----
#pragma unroll
    for (int r = 0; r < 16; ++r) {
      const float ad = (r & 1) ? aC[r >> 1].y : aC[r >> 1].x;
      const float kd = (r & 1) ? kC[r >> 1].y : kC[r >> 1].x;
      const float dl = fmaf(bw, kd, (ad - 1.0f) * sr[r]);
      const float s0 = S[0][r], s1 = S[1][r], s2 = S[2][r], s3 = S[3][r];
#pragma unroll
      for (int i = 0; i < 4; ++i) {
        float n = M[i * 4 + 0] * s0;
        n = fmaf(M[i * 4 + 1], s1, n);
        n = fmaf(M[i * 4 + 2], s2, n);
        n = fmaf(M[i * 4 + 3], s3, n);
        S[i][r] = fmaf(Hpost[i], dl, n);
      }
    }

    // ---- rotate pipeline registers ----
#pragma unroll
    for (int m = 0; m < 8; ++m) {
      kC[m] = kNx[m];
      aC[m] = aNx[m];
      qC[m] = qNx[m];
    }
    vlC = vlN; btC = btN;
  }
}

extern "C" void kernel_launch(void* const* d_in, const int* in_sizes, int n_in,
                              void* d_out, int out_size, void* d_ws, size_t ws_size,
                              hipStream_t stream) {
  (void)in_sizes; (void)n_in; (void)out_size; (void)d_ws; (void)ws_size;
  const float* q      = (const float*)d_in[0];
  const float* k      = (const float*)d_in[1];
  const float* v      = (const float*)d_in[2];
  const float* alpha  = (const float*)d_in[3];
  const float* beta   = (const float*)d_in[4];
  const float* S_init = (const float*)d_in[5];
  const float* w_norm = (const float*)d_in[6];
  const float* w_pre  = (const float*)d_in[7];
  const float* w_post = (const float*)d_in[8];
  const float* w_res  = (const float*)d_in[9];
  const float* b_pre  = (const float*)d_in[10];
  const float* b_post = (const float*)d_in[11];
  const float* b_res  = (const float*)d_in[12];
  const float* a_pre  = (const float*)d_in[13];
  const float* a_post = (const float*)d_in[14];
  const float* a_res  = (const float*)d_in[15];

  kda_mhc_wave_kernel<<<dim3(64), dim3(32), 0, stream>>>(
      q, k, v, alpha, beta, S_init, w_norm, w_pre, w_post, w_res,
      b_pre, b_post, b_res, a_pre, a_post, a_res, (float*)d_out);
}